// DynamicKeypathFusionModel_50886772523097
// MI455X (gfx1250) — compile-verified
//
#include <hip/hip_runtime.h>
#include <hip/hip_bf16.h>
#include <math.h>

// Problem constants (from reference)
constexpr int DIMC  = 128;
constexpr int NRELC = 16;

typedef __attribute__((ext_vector_type(2))) float v2f;
typedef __attribute__((ext_vector_type(8))) float v8f;

// ---------------------------------------------------------------------------
// h = entity_emb[node_ids]   (float4-vectorized gather)
// ---------------------------------------------------------------------------
__global__ void gather_rows_kernel(const float* __restrict__ emb,
                                   const int* __restrict__ ids,
                                   float* __restrict__ h, int n4) {
  int i = blockIdx.x * blockDim.x + threadIdx.x;
  if (i < n4) {
    int node = i >> 5;       // DIM/4 = 32 float4 per row
    int c4   = i & 31;
    ((float4*)h)[i] = ((const float4*)emb)[(size_t)ids[node] * 32 + c4];
  }
}

__global__ void zero_u32_kernel(unsigned* __restrict__ p, int n) {
  int i = blockIdx.x * blockDim.x + threadIdx.x;
  if (i < n) p[i] = 0u;
}

// per-(dst, relation) edge counts (RGCN 1/c_{i,r} normalizer) -- graph-static
__global__ void count_edges_kernel(const int* __restrict__ dst,
                                   const int* __restrict__ etype,
                                   unsigned* __restrict__ cnt, int e) {
  int i = blockIdx.x * blockDim.x + threadIdx.x;
  if (i < e) atomicAdd(&cnt[dst[i] * NRELC + etype[i]], 1u);
}

// ---------------------------------------------------------------------------
// Core 128-row-chunk GEMM tile: C[M,128] = A[M,128] @ B[128,128] using
// V_WMMA_F32_16X16X4_F32 (fp32 matrix pipe, preserves reference numerics).
// Block = 256 threads = 8 waves. A chunk + B staged in LDS with pad strides
// chosen for conflict-free fragment reads (APAD%64==4 spreads the 16 A rows;
// 2*BPAD%64==16 separates the two half-waves). Two interleaved accumulators
// break the WMMA RAW chain so each wave issues 2 independent XDL streams.
// ---------------------------------------------------------------------------
__device__ __forceinline__ void gemm_chunk_128(const float* __restrict__ A,
                                               const float* __restrict__ B,
                                               float* __restrict__ C, int M,
                                               int rowBlock, int tid,
                                               float* As, float* Bs) {
  constexpr int APAD = 132;
  constexpr int BPAD = 136;

  {
    const float4* B4 = (const float4*)B;
    const float4* A4 = (const float4*)A;
    float4 z; z.x = z.y = z.z = z.w = 0.f;
#pragma unroll
    for (int i = 0; i < 16; ++i) {
      int idx4 = tid + 256 * i;        // 0..4095 float4s
      int r    = idx4 >> 5;            // row 0..127
      int c4   = idx4 & 31;            // float4 column
      *(float4*)&Bs[r * BPAD + c4 * 4] = B4[idx4];
      int rg = rowBlock + r;
      float4 va = (rg < M) ? A4[(size_t)rg * 32 + c4] : z;
      *(float4*)&As[r * APAD + c4 * 4] = va;
    }
  }
  __syncthreads();

  const int lane    = tid & 31;
  const int hi      = lane >> 4;            // half-wave: selects K pair
  const int l16     = lane & 15;
  const int colBase = (tid >> 5) * 16;      // wave -> 16-column tile

#pragma unroll 1
  for (int m = 0; m < 8; ++m) {
    v8f acc0 = {};
    v8f acc1 = {};
    // A frag: lanes 0-15 rows at K=k0,k0+1; lanes 16-31 same rows K=k0+2,k0+3
    const float* arow = &As[(m * 16 + l16) * APAD + 2 * hi];
    // B frag: VGPR0 = row K (lanes 0-15) / K+2 (lanes 16-31); VGPR1 = K+1/K+3
    const float* bcol = &Bs[(2 * hi) * BPAD + colBase + l16];
#pragma unroll
    for (int k0 = 0; k0 < DIMC; k0 += 8) {
      v2f a0 = *(const v2f*)(arow + k0);
      v2f b0;
      b0.x = bcol[(k0) * BPAD];
      b0.y = bcol[(k0 + 1) * BPAD];
      acc0 = __builtin_amdgcn_wmma_f32_16x16x4_f32(false, a0, false, b0,
                                                   (short)0, acc0, false, false);
      v2f a1 = *(const v2f*)(arow + k0 + 4);
      v2f b1;
      b1.x = bcol[(k0 + 4) * BPAD];
      b1.y = bcol[(k0 + 5) * BPAD];
      acc1 = __builtin_amdgcn_wmma_f32_16x16x4_f32(false, a1, false, b1,
                                                   (short)0, acc1, false, false);
    }
    v8f accs = acc0 + acc1;
    // C/D layout: VGPR v -> row v (lanes 0-15) / row v+8 (lanes 16-31)
    const int rbase = rowBlock + m * 16 + hi * 8;
    const int col   = colBase + l16;
#pragma unroll
    for (int v = 0; v < 8; ++v) {
      int r = rbase + v;
      if (r < M) C[(size_t)r * DIMC + col] = accs[v];
    }
  }
}

// Single GEMM (fallback path / per-relation loop)
__global__ __launch_bounds__(256) void rgcn_gemm_wmma_kernel(
    const float* __restrict__ A, const float* __restrict__ B,
    float* __restrict__ C, int M) {
  __shared__ float As[DIMC * 132];
  __shared__ float Bs[DIMC * 136];
  gemm_chunk_128(A, B, C, M, blockIdx.x * 128, threadIdx.x, As, Bs);
}

// Batched GEMM: blockIdx.y == 0 -> acc = h @ W_self[l]
//               blockIdx.y == r+1 -> hrel[r] = h @ W_rel[l][r]
__global__ __launch_bounds__(256) void rgcn_gemm_batch_kernel(
    const float* __restrict__ A, const float* __restrict__ Wself,
    const float* __restrict__ Wrel, float* __restrict__ acc,
    float* __restrict__ hrel, int M) {
  __shared__ float As[DIMC * 132];
  __shared__ float Bs[DIMC * 136];
  const int y = blockIdx.y;
  const float* B = (y == 0) ? Wself : (Wrel + (size_t)(y - 1) * DIMC * DIMC);
  float* C = (y == 0) ? acc : (hrel + (size_t)(y - 1) * M * DIMC);
  gemm_chunk_128(A, B, C, M, blockIdx.x * 128, threadIdx.x, As, Bs);
}

// ---------------------------------------------------------------------------
// Batched scatter (single pass over all edges):
//   acc[dst] += hrel[etype][src] / max(cnt[dst,etype],1)
// One wave per edge, 4 floats per lane, f32 global atomics (acc L2-resident).
// ---------------------------------------------------------------------------
__global__ __launch_bounds__(256) void scatter_all_kernel(
    const int* __restrict__ src, const int* __restrict__ dst,
    const int* __restrict__ etype, const unsigned* __restrict__ cnt,
    const float* __restrict__ hrel, float* __restrict__ acc, int e, int n) {
  int wave   = (blockIdx.x * blockDim.x + threadIdx.x) >> 5;
  int lane   = threadIdx.x & 31;
  int nwaves = (gridDim.x * blockDim.x) >> 5;
  for (int ed = wave; ed < e; ed += nwaves) {
    int r = etype[ed];
    int d = dst[ed], s = src[ed];
    unsigned c = cnt[d * NRELC + r];
    float inv = 1.0f / (float)(c > 1u ? c : 1u);
    const float* tp = hrel + ((size_t)r * n + s) * DIMC;
    float* ap = acc + (size_t)d * DIMC;
#pragma unroll
    for (int j = 0; j < 4; ++j) {
      atomicAdd(&ap[lane + 32 * j], tp[lane + 32 * j] * inv);
    }
  }
}

// Per-relation scatter (fallback path; T stays L2-resident)
__global__ __launch_bounds__(256) void scatter_rel_kernel(
    const int* __restrict__ src, const int* __restrict__ dst,
    const int* __restrict__ etype, const unsigned* __restrict__ cnt,
    const float* __restrict__ T, float* __restrict__ acc, int e, int r) {
  int wave   = (blockIdx.x * blockDim.x + threadIdx.x) >> 5;
  int lane   = threadIdx.x & 31;
  int nwaves = (gridDim.x * blockDim.x) >> 5;
  for (int ed = wave; ed < e; ed += nwaves) {
    if (etype[ed] != r) continue;
    int d = dst[ed], s = src[ed];
    unsigned c = cnt[d * NRELC + r];
    float inv = 1.0f / (float)(c > 1u ? c : 1u);
    const float* tp = T + (size_t)s * DIMC;
    float* ap = acc + (size_t)d * DIMC;
#pragma unroll
    for (int j = 0; j < 4; ++j) {
      atomicAdd(&ap[lane + 32 * j], tp[lane + 32 * j] * inv);
    }
  }
}

__global__ void relu4_kernel(const float* __restrict__ in,
                             float* __restrict__ out, int n4) {
  int i = blockIdx.x * blockDim.x + threadIdx.x;
  if (i < n4) {
    float4 v = ((const float4*)in)[i];
    v.x = fmaxf(v.x, 0.f); v.y = fmaxf(v.y, 0.f);
    v.z = fmaxf(v.z, 0.f); v.w = fmaxf(v.w, 0.f);
    ((float4*)out)[i] = v;
  }
}

// ---------------------------------------------------------------------------
// DistMult score + path branch; one wave per query, wave32 shuffle reduction.
// ---------------------------------------------------------------------------
__global__ __launch_bounds__(128) void score_kernel(
    const float* __restrict__ h, const int* __restrict__ heads,
    const int* __restrict__ rels, const int* __restrict__ tails,
    const float* __restrict__ rel_emb, const float* __restrict__ path_feat,
    const int* __restrict__ rel_task_idx, const float* __restrict__ delta_w,
    const float* __restrict__ lambda_logit, const float* __restrict__ rule_init,
    float* __restrict__ out, int q) {
  int wave = (blockIdx.x * blockDim.x + threadIdx.x) >> 5;
  int lane = threadIdx.x & 31;
  if (wave >= q) return;
  const float* hp = h + (size_t)heads[wave] * DIMC;
  const float* tp = h + (size_t)tails[wave] * DIMC;
  const float* rp = rel_emb + (size_t)rels[wave] * DIMC;
  float s = 0.f;
#pragma unroll
  for (int j = 0; j < 4; ++j) {
    int d = lane + 32 * j;
    s += hp[d] * rp[d] * tp[d];
  }
  for (int off = 16; off; off >>= 1) s += __shfl_xor(s, off, 32);
  if (lane == 0) {
    int idx = rel_task_idx[0];
    float sp = 0.f;
#pragma unroll
    for (int j = 0; j < 5; ++j)
      sp += path_feat[(size_t)wave * 5 + j] *
            (rule_init[idx * 5 + j] + delta_w[idx * 5 + j]);
    float lam = 1.f / (1.f + __expf(-lambda_logit[idx]));
    out[wave] = lam * s + (1.f - lam) * sp;
  }
}

// ---------------------------------------------------------------------------
extern "C" void kernel_launch(void* const* d_in, const int* in_sizes, int n_in,
                              void* d_out, int out_size, void* d_ws, size_t ws_size,
                              hipStream_t stream) {
  const int*   node_ids   = (const int*)d_in[0];
  const int*   edge_index = (const int*)d_in[1];
  const int*   etype      = (const int*)d_in[2];
  const int*   heads      = (const int*)d_in[3];
  const int*   rels       = (const int*)d_in[4];
  const int*   tails      = (const int*)d_in[5];
  const float* path_feat  = (const float*)d_in[6];
  const int*   rel_task   = (const int*)d_in[7];
  const float* entity_emb = (const float*)d_in[8];
  const float* rel_emb    = (const float*)d_in[9];
  const float* W_self     = (const float*)d_in[10];
  const float* W_rel      = (const float*)d_in[11];
  const float* delta_w    = (const float*)d_in[12];
  const float* lambda_l   = (const float*)d_in[13];
  const float* rule_init  = (const float*)d_in[14];

  const int N = in_sizes[0];            // 50000
  const int E = in_sizes[2];            // 600000
  const int Q = in_sizes[3];            // 8192
  const int* srcp = edge_index;         // edge_index[0,:]
  const int* dstp = edge_index + E;     // edge_index[1,:]

  const size_t nd = (size_t)N * DIMC;
  // Batched path workspace: hA, hB, acc, hrel[16], cnt
  const size_t need_big = (3 * nd + (size_t)NRELC * nd + (size_t)N * NRELC) * 4;
  const bool big = (ws_size >= need_big);

  float*    hA  = (float*)d_ws;
  float*    hB  = hA + nd;
  float*    acc = hB + nd;
  float*    big_hrel = acc + nd;                       // batched path only
  unsigned* big_cnt  = (unsigned*)(big_hrel + (size_t)NRELC * nd);
  float*    sm_T     = acc + nd;                       // fallback path only
  unsigned* sm_cnt   = (unsigned*)(sm_T + nd);
  unsigned* cnt = big ? big_cnt : sm_cnt;

  const int n4 = N * (DIMC / 4);
  gather_rows_kernel<<<(n4 + 255) / 256, 256, 0, stream>>>(entity_emb, node_ids, hA, n4);
  zero_u32_kernel<<<(N * NRELC + 255) / 256, 256, 0, stream>>>(cnt, N * NRELC);
  count_edges_kernel<<<(E + 255) / 256, 256, 0, stream>>>(dstp, etype, cnt, E);

  const int gemmGridX = (N + 127) / 128;
  float* hcur = hA;
  float* hnxt = hB;
  for (int l = 0; l < 2; ++l) {
    const float* Wself_l = W_self + (size_t)l * DIMC * DIMC;
    const float* Wrel_l  = W_rel + (size_t)l * NRELC * DIMC * DIMC;
    if (big) {
      // One launch: acc = h@Ws and hrel[r] = h@Wr for all 16 relations
      dim3 grid(gemmGridX, NRELC + 1);
      rgcn_gemm_batch_kernel<<<grid, 256, 0, stream>>>(
          hcur, Wself_l, Wrel_l, acc, big_hrel, N);
      // One edge pass: acc[dst] += hrel[etype][src] / c_{dst,etype}
      scatter_all_kernel<<<2048, 256, 0, stream>>>(
          srcp, dstp, etype, cnt, big_hrel, acc, E, N);
    } else {
      rgcn_gemm_wmma_kernel<<<gemmGridX, 256, 0, stream>>>(hcur, Wself_l, acc, N);
      for (int r = 0; r < NRELC; ++r) {
        rgcn_gemm_wmma_kernel<<<gemmGridX, 256, 0, stream>>>(
            hcur, Wrel_l + (size_t)r * DIMC * DIMC, sm_T, N);
        scatter_rel_kernel<<<1024, 256, 0, stream>>>(
            srcp, dstp, etype, cnt, sm_T, acc, E, r);
      }
    }
    relu4_kernel<<<(n4 + 255) / 256, 256, 0, stream>>>(acc, hnxt, n4);
    float* t = hcur; hcur = hnxt; hnxt = t;
  }

  score_kernel<<<(Q * 32 + 127) / 128, 128, 0, stream>>>(
      hcur, heads, rels, tails, rel_emb, path_feat, rel_task,
      delta_w, lambda_l, rule_init, (float*)d_out, Q);
}